// RelationalGATConv_83056077570509
// MI455X (gfx1250) — compile-verified
//
#include <hip/hip_runtime.h>
#include <hip/hip_bf16.h>
#include <math.h>

typedef __attribute__((ext_vector_type(2))) float v2f;
typedef __attribute__((ext_vector_type(8))) float v8f;
typedef int v4i __attribute__((vector_size(16)));
typedef __attribute__((address_space(1))) v4i* gptr_v4i;
typedef __attribute__((address_space(3))) v4i* lptr_v4i;

#define IN_F 128
#define OUT_F 32
#define H 4
#define R 8
#define NCOLS 192   // 128 (ft) + 32 (al) + 32 (ar)
#define SLOPE 0.01f

#if defined(__gfx1250__) && __has_builtin(__builtin_amdgcn_global_load_async_to_lds_b128)
#define HAVE_ASYNC_LDS 1
#else
#define HAVE_ASYNC_LDS 0
#endif

__device__ __forceinline__ float leaky(float x) { return x >= 0.0f ? x : SLOPE * x; }

// ---------------------------------------------------------------------------
// 1) Build combined B matrix [128 x 192] and bias [192]:
//    cols 0..127  : fc_W
//    cols 128..159: vl[j][k] = sum_o Wl[j,k,o] * Wo[j,o]        (j = h*8+r)
//    cols 160..191: vr[j][k] = sum_o Wr[j,k,o] * Wo[j,32+o]
//    bias = { fc_b, cl, cr }
// ---------------------------------------------------------------------------
__global__ void precompute_kernel(const float* __restrict__ fc_W,
                                  const float* __restrict__ fc_b,
                                  const float* __restrict__ Wl,
                                  const float* __restrict__ bl,
                                  const float* __restrict__ Wr,
                                  const float* __restrict__ br,
                                  const float* __restrict__ Wo,
                                  float* __restrict__ Bmat,
                                  float* __restrict__ bias) {
    int t = blockIdx.x * blockDim.x + threadIdx.x;
    if (t >= IN_F * NCOLS) return;
    int col = t % NCOLS;
    int k   = t / NCOLS;
    float v;
    if (col < 128) {
        v = fc_W[k * 128 + col];
    } else if (col < 160) {
        int j = col - 128;
        const float* w  = Wl + (size_t)(j * IN_F + k) * OUT_F;
        const float* wo = Wo + j * (2 * OUT_F);
        float s = 0.f;
        #pragma unroll
        for (int o = 0; o < OUT_F; ++o) s += w[o] * wo[o];
        v = s;
    } else {
        int j = col - 160;
        const float* w  = Wr + (size_t)(j * IN_F + k) * OUT_F;
        const float* wo = Wo + j * (2 * OUT_F) + OUT_F;
        float s = 0.f;
        #pragma unroll
        for (int o = 0; o < OUT_F; ++o) s += w[o] * wo[o];
        v = s;
    }
    Bmat[k * NCOLS + col] = v;
    if (k == 0) {
        float bvv;
        if (col < 128) {
            bvv = fc_b[col];
        } else if (col < 160) {
            int j = col - 128; float s = 0.f;
            #pragma unroll
            for (int o = 0; o < OUT_F; ++o) s += bl[j * OUT_F + o] * Wo[j * 64 + o];
            bvv = s;
        } else {
            int j = col - 160; float s = 0.f;
            #pragma unroll
            for (int o = 0; o < OUT_F; ++o) s += br[j * OUT_F + o] * Wo[j * 64 + 32 + o];
            bvv = s;
        }
        bias[col] = bvv;
    }
}

// ---------------------------------------------------------------------------
// 2) zero the CSR count / cursor arrays
// ---------------------------------------------------------------------------
__global__ void init_kernel(int* __restrict__ cnt, int* __restrict__ cur, int n) {
    int i = blockIdx.x * blockDim.x + threadIdx.x;
    if (i < n) { cnt[i] = 0; cur[i] = 0; }
}

// ---------------------------------------------------------------------------
// 3) Fused node GEMM via V_WMMA_F32_16X16X4_F32:
//    [N,128] x [128,192] + bias -> ft [N,128], al [N,32], ar [N,32]
//    Block = 384 threads = 12 waves; wave w computes 16x16 tile of columns
//    w*16..w*16+15 for the block's 16 rows. A tile staged in padded LDS,
//    via the CDNA5 async global->LDS path when available.
// ---------------------------------------------------------------------------
__global__ __launch_bounds__(384) void gemm_node_kernel(
    const float* __restrict__ feat, const float* __restrict__ Bmat,
    const float* __restrict__ bias, float* __restrict__ ft,
    float* __restrict__ al, float* __restrict__ ar, int n_nodes) {
    __shared__ float As[16 * 132];        // +4 pad: stride 132 kills bank conflicts
    const int tileM = blockIdx.x;
    const int tid   = threadIdx.x;
    const int row0  = tileM * 16;

#if HAVE_ASYNC_LDS
    // 512 x 16B chunks; chunks never straddle the padded rows (132*4 B is 16B-aligned)
    for (int c = tid; c < 512; c += 384) {
        int r = c >> 5, col4 = (c & 31) * 4;
        int grow = row0 + r;
        if (grow < n_nodes) {
            __builtin_amdgcn_global_load_async_to_lds_b128(
                (gptr_v4i)(feat + (size_t)grow * IN_F + col4),
                (lptr_v4i)(&As[r * 132 + col4]),
                /*offset=*/0, /*cpol=*/0);
        } else {
            As[r * 132 + col4 + 0] = 0.f; As[r * 132 + col4 + 1] = 0.f;
            As[r * 132 + col4 + 2] = 0.f; As[r * 132 + col4 + 3] = 0.f;
        }
    }
#if __has_builtin(__builtin_amdgcn_s_wait_asynccnt)
    __builtin_amdgcn_s_wait_asynccnt(0);
#else
    asm volatile("s_wait_asynccnt 0x0" ::: "memory");
#endif
#else
    for (int i = tid; i < 16 * 128; i += 384) {
        int r = i >> 7, c = i & 127;
        int grow = row0 + r;
        As[r * 132 + c] = (grow < n_nodes) ? feat[(size_t)grow * IN_F + c] : 0.0f;
    }
#endif
    __syncthreads();

    const int wave = tid >> 5;            // 0..11 -> column tile
    const int lane = tid & 31;
    const int m    = lane & 15;           // A row within tile
    const int kb   = (lane >> 4) << 1;    // 0 or 2
    const int colg = wave * 16 + (lane & 15);

    v8f c;
    {
        float bv = bias[colg];
        #pragma unroll
        for (int v = 0; v < 8; ++v) c[v] = bv;
    }

    #pragma unroll 4
    for (int k = 0; k < IN_F; k += 4) {
        v2f a, b;
        a.x = As[m * 132 + k + kb];
        a.y = As[m * 132 + k + kb + 1];
        b.x = Bmat[(size_t)(k + kb)     * NCOLS + colg];
        b.y = Bmat[(size_t)(k + kb + 1) * NCOLS + colg];
        c = __builtin_amdgcn_wmma_f32_16x16x4_f32(
                /*neg_a=*/false, a, /*neg_b=*/false, b,
                /*c_mod=*/(short)0, c, /*reuse_a=*/false, /*reuse_b=*/false);
    }

    const int mbase = (lane >> 4) << 3;   // C/D: vgpr v holds row v + (lane>=16)*8
    #pragma unroll
    for (int v = 0; v < 8; ++v) {
        int grow = row0 + mbase + v;
        if (grow >= n_nodes) continue;
        float val = c[v];
        if (colg < 128)       ft[(size_t)grow * 128 + colg]         = val;
        else if (colg < 160)  al[(size_t)grow * 32  + (colg - 128)] = val;
        else                  ar[(size_t)grow * 32  + (colg - 160)] = val;
    }
}

// ---------------------------------------------------------------------------
// 4) CSR build: count, exclusive scan (3 kernels), fill
// ---------------------------------------------------------------------------
__global__ void count_kernel(const int* __restrict__ dst, int* __restrict__ cnt, int E) {
    int e = blockIdx.x * blockDim.x + threadIdx.x;
    if (e < E) atomicAdd(&cnt[dst[e]], 1);
}

#define SCAN_B 256
__global__ void scan_block_kernel(const int* __restrict__ cnt, int* __restrict__ loc,
                                  int* __restrict__ bsums, int n) {
    __shared__ int sh[SCAN_B];
    int i = blockIdx.x * SCAN_B + threadIdx.x;
    int v = (i < n) ? cnt[i] : 0;
    sh[threadIdx.x] = v;
    __syncthreads();
    for (int d = 1; d < SCAN_B; d <<= 1) {
        int t = (threadIdx.x >= d) ? sh[threadIdx.x - d] : 0;
        __syncthreads();
        sh[threadIdx.x] += t;
        __syncthreads();
    }
    if (i < n) loc[i] = sh[threadIdx.x] - v;                 // exclusive within block
    if (threadIdx.x == SCAN_B - 1) bsums[blockIdx.x] = sh[SCAN_B - 1];
}

__global__ void scan_sums_kernel(int* __restrict__ bsums, int nb) {
    __shared__ int sh[1024];
    int t = threadIdx.x;
    int v = (t < nb) ? bsums[t] : 0;
    sh[t] = v;
    __syncthreads();
    for (int d = 1; d < 1024; d <<= 1) {
        int x = (t >= d) ? sh[t - d] : 0;
        __syncthreads();
        sh[t] += x;
        __syncthreads();
    }
    if (t < nb) bsums[t] = sh[t] - v;                        // exclusive block offsets
}

__global__ void scan_finalize_kernel(const int* __restrict__ loc, const int* __restrict__ bsums,
                                     const int* __restrict__ cnt, int* __restrict__ off, int n) {
    int i = blockIdx.x * SCAN_B + threadIdx.x;
    if (i < n) {
        int o = loc[i] + bsums[i >> 8];
        off[i] = o;
        if (i == n - 1) off[n] = o + cnt[i];                 // == E
    }
}

__global__ void fill_kernel(const int* __restrict__ dst, const int* __restrict__ off,
                            int* __restrict__ cur, int* __restrict__ eidx, int E) {
    int e = blockIdx.x * blockDim.x + threadIdx.x;
    if (e >= E) return;
    int d = dst[e];
    int p = off[d] + atomicAdd(&cur[d], 1);
    eidx[p] = e;
}

// ---------------------------------------------------------------------------
// 5) One wave per node: fused softmax + weighted aggregation, no atomics.
//    P1: per-head max (lanes stripe edges, shfl_xor reduce)
//    P2: per-head sum of exp(e - mx)
//    P3: edges serial (wave-uniform), lanes own 4 of 128 output features.
// ---------------------------------------------------------------------------
__global__ __launch_bounds__(256) void node_aggregate_kernel(
    const int* __restrict__ off, const int* __restrict__ eidx,
    const int* __restrict__ src, const int* __restrict__ et,
    const float* __restrict__ al, const float* __restrict__ ar,
    const float* __restrict__ bo, const float* __restrict__ ft,
    float* __restrict__ out, int n_nodes) {
    int n = (blockIdx.x * blockDim.x + threadIdx.x) >> 5;
    int lane = threadIdx.x & 31;
    if (n >= n_nodes) return;
    int beg = off[n], end = off[n + 1];

    float mxh[H] = {-INFINITY, -INFINITY, -INFINITY, -INFINITY};
    for (int i = beg + lane; i < end; i += 32) {
        int e = eidx[i]; int s = src[e]; int r = et[e];
        #pragma unroll
        for (int h = 0; h < H; ++h) {
            int j = h * R + r;
            float ev = leaky(al[(size_t)s * 32 + j] + ar[(size_t)n * 32 + j] + bo[j]);
            mxh[h] = fmaxf(mxh[h], ev);
        }
    }
    #pragma unroll
    for (int m = 16; m >= 1; m >>= 1) {
        #pragma unroll
        for (int h = 0; h < H; ++h) mxh[h] = fmaxf(mxh[h], __shfl_xor(mxh[h], m, 32));
    }

    float dnh[H] = {0.f, 0.f, 0.f, 0.f};
    for (int i = beg + lane; i < end; i += 32) {
        int e = eidx[i]; int s = src[e]; int r = et[e];
        #pragma unroll
        for (int h = 0; h < H; ++h) {
            int j = h * R + r;
            float ev = leaky(al[(size_t)s * 32 + j] + ar[(size_t)n * 32 + j] + bo[j]);
            dnh[h] += __expf(ev - mxh[h]);
        }
    }
    #pragma unroll
    for (int m = 16; m >= 1; m >>= 1) {
        #pragma unroll
        for (int h = 0; h < H; ++h) dnh[h] += __shfl_xor(dnh[h], m, 32);
    }

    // this lane's head (avoid dynamic array index -> keep in VGPRs)
    int h = lane >> 3;
    float mxv = (h & 2) ? ((h & 1) ? mxh[3] : mxh[2]) : ((h & 1) ? mxh[1] : mxh[0]);
    float dnv = (h & 2) ? ((h & 1) ? dnh[3] : dnh[2]) : ((h & 1) ? dnh[1] : dnh[0]);
    float invd = (dnv > 0.f) ? 1.0f / dnv : 0.f;

    float4 acc = make_float4(0.f, 0.f, 0.f, 0.f);
    for (int i = beg; i < end; ++i) {       // wave-uniform edge loop
        int e = eidx[i]; int s = src[e]; int r = et[e];
        int j = h * R + r;
        float ev = leaky(al[(size_t)s * 32 + j] + ar[(size_t)n * 32 + j] + bo[j]);
        float a = __expf(ev - mxv) * invd;
        const float4 f = *(const float4*)(ft + (size_t)s * 128 + lane * 4);
        acc.x += a * f.x; acc.y += a * f.y; acc.z += a * f.z; acc.w += a * f.w;
    }
    float4 o = make_float4(leaky(acc.x), leaky(acc.y), leaky(acc.z), leaky(acc.w));
    *(float4*)(out + (size_t)n * 128 + lane * 4) = o;
}

extern "C" void kernel_launch(void* const* d_in, const int* in_sizes, int n_in,
                              void* d_out, int out_size, void* d_ws, size_t ws_size,
                              hipStream_t stream) {
    const float* feat = (const float*)d_in[0];
    const int*   src  = (const int*)d_in[1];
    const int*   dst  = (const int*)d_in[2];
    const int*   et   = (const int*)d_in[3];
    const float* fc_W = (const float*)d_in[4];
    const float* fc_b = (const float*)d_in[5];
    const float* Wl   = (const float*)d_in[6];
    const float* bl   = (const float*)d_in[7];
    const float* Wr   = (const float*)d_in[8];
    const float* br   = (const float*)d_in[9];
    const float* Wo   = (const float*)d_in[10];
    const float* bo   = (const float*)d_in[11];

    const int N = in_sizes[0] / IN_F;   // 50000
    const int E = in_sizes[1];          // 800000
    float* out = (float*)d_out;

    // workspace layout
    float* ws    = (float*)d_ws;
    float* Bmat  = ws;                          // 128*192
    float* bias  = Bmat + IN_F * NCOLS;         // 192
    float* al    = bias + NCOLS;                // N*32
    float* ar    = al + (size_t)N * 32;         // N*32
    float* ft    = ar + (size_t)N * 32;         // N*128
    int*   icnt  = (int*)(ft + (size_t)N * 128);// N
    int*   iloc  = icnt + N;                    // N
    int*   ibsum = iloc + N;                    // 1024
    int*   ioff  = ibsum + 1024;                // N+1
    int*   icur  = ioff + (N + 1);              // N
    int*   ieidx = icur + N;                    // E

    const int nb = (N + SCAN_B - 1) / SCAN_B;   // 196 <= 1024

    precompute_kernel<<<(IN_F * NCOLS + 255) / 256, 256, 0, stream>>>(
        fc_W, fc_b, Wl, bl, Wr, br, Wo, Bmat, bias);

    init_kernel<<<(N + 255) / 256, 256, 0, stream>>>(icnt, icur, N);

    gemm_node_kernel<<<(N + 15) / 16, 384, 0, stream>>>(feat, Bmat, bias, ft, al, ar, N);

    count_kernel<<<(E + 255) / 256, 256, 0, stream>>>(dst, icnt, E);
    scan_block_kernel<<<nb, SCAN_B, 0, stream>>>(icnt, iloc, ibsum, N);
    scan_sums_kernel<<<1, 1024, 0, stream>>>(ibsum, nb);
    scan_finalize_kernel<<<nb, SCAN_B, 0, stream>>>(iloc, ibsum, icnt, ioff, N);
    fill_kernel<<<(E + 255) / 256, 256, 0, stream>>>(dst, ioff, icur, ieidx, E);

    node_aggregate_kernel<<<(N * 32 + 255) / 256, 256, 0, stream>>>(
        ioff, ieidx, src, et, al, ar, bo, ft, out, N);
}